// maxpool2d_23545010716727
// MI455X (gfx1250) — compile-verified
//
#include <hip/hip_runtime.h>

// Hexagonal max-pool (SIZE=2, STRIDE=1) on (16,64,256,256) fp32.
// Equivalent single-pass stencil per output (h,w), OOB taps = 0.0f:
//   max( in[h-2..h+2, w],
//        in[h-2..h+1, w+-1]  (w even)  /  in[h-1..h+2, w+-1] (w odd),
//        in[h-1..h+1, w+-2] )
// Bandwidth-bound (~536 MB moved, ~25 us at 23.3 TB/s). Strategy:
//  - stage 68-row x 256-col strips (2-row/2-col halo) in LDS via CDNA5
//    async global->LDS B128 loads (ASYNCcnt path),
//  - one column per lane, rolling 5-deep vertical max windows in registers,
//  - non-temporal coalesced stores (output is never re-read; keep L2 for input).

#define TILE_H   64
#define HALO     2
#define LDS_H    (TILE_H + 2 * HALO)   // 68 rows
#define LDS_W    264                   // 4 pad | 256 interior | 4 pad (16B-aligned interior)
#define THREADS  256
#define STRIPS   4                     // 256 / TILE_H (power of two)

typedef int v4i __attribute__((ext_vector_type(4)));
typedef __attribute__((address_space(1))) v4i  g_v4i;   // global (HIP prints as __device__)
typedef __attribute__((address_space(3))) v4i  l_v4i;   // LDS
typedef __attribute__((address_space(3))) void lds_void;

__device__ __forceinline__ void async_copy_b128_to_lds(const void* gptr, void* lptr) {
#if defined(__gfx1250__) && __has_builtin(__builtin_amdgcn_global_load_async_to_lds_b128)
    __builtin_amdgcn_global_load_async_to_lds_b128((g_v4i*)gptr, (l_v4i*)lptr, 0, 0);
#else
    // vdst = LDS byte offset (AS3 pointers are 32-bit offsets), vaddr = 64-bit global addr
    unsigned loff = (unsigned)(unsigned long long)(lds_void*)lptr;
    asm volatile("global_load_async_to_lds_b128 %0, %1, off"
                 :: "v"(loff), "v"(gptr) : "memory");
#endif
}

__device__ __forceinline__ void wait_async_zero() {
#if defined(__gfx1250__) && __has_builtin(__builtin_amdgcn_s_wait_asynccnt)
    __builtin_amdgcn_s_wait_asynccnt(0);
#else
    asm volatile("s_wait_asynccnt 0" ::: "memory");
#endif
}

__global__ __launch_bounds__(THREADS)
void hex_maxpool_s2s1_kernel(const float* __restrict__ in,
                             float* __restrict__ out,
                             int H, int W) {
    __shared__ float tile[LDS_H * LDS_W];

    const int t     = threadIdx.x;
    const int img   = (int)blockIdx.x >> 2;        // blockIdx / STRIPS
    const int strip = (int)blockIdx.x & (STRIPS - 1);
    const int h0    = strip * TILE_H;
    const size_t imgBase = (size_t)img * (size_t)H * (size_t)W;

    // Zero the 4 pad columns on each side (reference pads with zeros, not -inf).
    for (int p = t; p < LDS_H * 8; p += THREADS) {
        const int r   = p >> 3;
        const int c   = p & 7;
        const int col = (c < 4) ? c : (LDS_W - 8 + c);
        tile[r * LDS_W + col] = 0.0f;
    }

    // Stage 68 rows x 256 cols into LDS interior as 16B chunks.
    // 68 * 64 = 4352 chunks, 17 per thread, coalesced across lanes.
#pragma unroll
    for (int k = 0; k < (LDS_H * 64) / THREADS; ++k) {
        const int chunk = t + THREADS * k;
        const int r   = chunk >> 6;        // LDS row 0..67
        const int c4  = chunk & 63;        // 16B chunk within row
        const int gr  = h0 - HALO + r;     // global row (may be OOB)
        float* lp = &tile[r * LDS_W + 4 + c4 * 4];
        if ((unsigned)gr < (unsigned)H) {
            const float* gp = in + imgBase + (size_t)gr * W + (size_t)(c4 * 4);
            async_copy_b128_to_lds(gp, lp);
        } else {
            *(float4*)lp = make_float4(0.f, 0.f, 0.f, 0.f);  // OOB rows contribute 0
        }
    }
    wait_async_zero();
    __syncthreads();

    // One output column per thread, rolling 5-deep windows down the strip.
    const int  w       = t;                // 0..255
    const int  cw      = 4 + w;            // LDS interior column
    const bool evenCol = ((w & 1) == 0);
    float* outp = out + imgBase + (size_t)h0 * W + (size_t)w;

    float A0=0.f,A1=0.f,A2=0.f,A3=0.f,A4=0.f;   // center column stream
    float B0=0.f,B1=0.f,B2=0.f,B3=0.f,B4=0.f;   // max(col-1, col+1) stream
    float C1=0.f,C2=0.f,C3=0.f,C4=0.f;          // max(col-2, col+2) stream

#pragma unroll
    for (int r = 0; r < LDS_H; ++r) {
        const float* rowp = &tile[r * LDS_W + cw];
        const float ctr = rowp[0];
        const float bn  = fmaxf(rowp[-1], rowp[1]);
        const float cn  = fmaxf(rowp[-2], rowp[2]);
        A0=A1; A1=A2; A2=A3; A3=A4; A4=ctr;
        B0=B1; B1=B2; B2=B3; B3=B4; B4=bn;
        C1=C2; C2=C3; C3=C4; C4=cn;
        if (r >= 2 * HALO) {
            // emitting local row ho = r-4; window rows ho..ho+4 == {A0..A4}
            const float ma = fmaxf(fmaxf(fmaxf(A0, A1), fmaxf(A2, A3)), A4);
            const float mb = evenCol ? fmaxf(fmaxf(B0, B1), fmaxf(B2, B3))   // rows h-2..h+1
                                     : fmaxf(fmaxf(B1, B2), fmaxf(B3, B4));  // rows h-1..h+2
            const float mc = fmaxf(fmaxf(C1, C2), C3);                       // rows h-1..h+1
            __builtin_nontemporal_store(fmaxf(ma, fmaxf(mb, mc)),
                                        outp + (size_t)(r - 2 * HALO) * W);
        }
    }
}

extern "C" void kernel_launch(void* const* d_in, const int* in_sizes, int n_in,
                              void* d_out, int out_size, void* d_ws, size_t ws_size,
                              hipStream_t stream) {
    const float* in  = (const float*)d_in[0];
    float*       out = (float*)d_out;

    const int N = 16, C = 64, H = 256, W = 256;   // reference setup_inputs shape
    const int blocks = N * C * STRIPS;            // 1024 images * 4 strips = 4096

    hex_maxpool_s2s1_kernel<<<blocks, THREADS, 0, stream>>>(in, out, H, W);
}